// JKNet_8134668058764
// MI455X (gfx1250) — compile-verified
//
#include <hip/hip_runtime.h>

#define HID 128
#define OUTD 64
#define NLAYERS 3

typedef float v2f __attribute__((ext_vector_type(2)));
typedef float v8f __attribute__((ext_vector_type(8)));

// D = A(16x4 f32) * B(4x16 f32) + C  -- full-precision CDNA5 WMMA
__device__ __forceinline__ v8f wmma_k4(v2f a, v2f b, v8f c) {
    return __builtin_amdgcn_wmma_f32_16x16x4_f32(
        /*neg_a=*/false, a, /*neg_b=*/false, b,
        /*c_mod=*/(short)0, c, /*reuse_a=*/false, /*reuse_b=*/false);
}

// ---------------- utility kernels ----------------

__global__ void k_zero(float* __restrict__ p, int n) {
    int i = blockIdx.x * blockDim.x + threadIdx.x;
    if (i < n) p[i] = 0.0f;
}

__global__ void k_zero4(float4* __restrict__ p, int n4) {
    int i = blockIdx.x * blockDim.x + threadIdx.x;
    if (i < n4) p[i] = make_float4(0.f, 0.f, 0.f, 0.f);
}

// out[n, j] = fc_b[j]   (OUTD == 64, power of two)
__global__ void k_init_out(float* __restrict__ out, const float* __restrict__ fcb, int total) {
    int i = blockIdx.x * blockDim.x + threadIdx.x;
    if (i < total) out[i] = fcb[i & (OUTD - 1)];
}

// deg[dst[e]] += 1   (400 KB array -> L2-resident atomics)
__global__ void k_degree(const int* __restrict__ dst, float* __restrict__ deg, int E) {
    int e = blockIdx.x * blockDim.x + threadIdx.x;
    if (e < E) unsafeAtomicAdd(&deg[dst[e]], 1.0f);
}

// agg[dst[e], :] += h[src[e], :]   one wave per edge, float4 per lane,
// HW float atomics into the L2-resident 51.2 MB aggregation buffer.
__global__ void k_scatter(const float* __restrict__ hin, const int* __restrict__ src,
                          const int* __restrict__ dst, float* __restrict__ agg, int E) {
    int gid = blockIdx.x * blockDim.x + threadIdx.x;
    int e = gid >> 5;
    if (e >= E) return;
    int lane = gid & 31;
    int s = src[e];
    int d = dst[e];
    const float4 v = *reinterpret_cast<const float4*>(hin + (size_t)s * HID + lane * 4);
    float* a = agg + (size_t)d * HID + lane * 4;
    unsafeAtomicAdd(a + 0, v.x);
    unsafeAtomicAdd(a + 1, v.y);
    unsafeAtomicAdd(a + 2, v.z);
    unsafeAtomicAdd(a + 3, v.w);
}

// h_out = relu( (agg/deg) @ Wl + h_in @ Wr + b )
// block = 256 (8 waves); block owns 16 nodes x 128 out; wave w owns col tile w.
__global__ void __launch_bounds__(256)
k_sage(const float* __restrict__ hin, const float* __restrict__ agg,
       const float* __restrict__ deg, const float* __restrict__ Wl,
       const float* __restrict__ Wr, const float* __restrict__ bias,
       float* __restrict__ hout) {
    const int w    = threadIdx.x >> 5;   // wave id 0..7 -> col tile
    const int lane = threadIdx.x & 31;
    const int half = lane >> 4;          // 0: K0/K1  1: K2/K3
    const int l15  = lane & 15;
    const int mrow = blockIdx.x * 16 + l15;
    const int nb   = w * 16;

    const float* hrow = hin + (size_t)mrow * HID + 2 * half;
    const float* arow = agg + (size_t)mrow * HID + 2 * half;
    const float  invd = 1.0f / fmaxf(deg[mrow], 1.0f);

    const float* blp = Wl + (size_t)(2 * half) * HID + nb + l15;
    const float* brp = Wr + (size_t)(2 * half) * HID + nb + l15;

    v8f c = {};
    for (int k0 = 0; k0 < HID; k0 += 4) {
        float2 am2 = *reinterpret_cast<const float2*>(arow + k0);
        float2 ah2 = *reinterpret_cast<const float2*>(hrow + k0);
        v2f am; am.x = am2.x * invd; am.y = am2.y * invd;
        v2f ah; ah.x = ah2.x;        ah.y = ah2.y;
        const float* bl = blp + (size_t)k0 * HID;
        const float* br = brp + (size_t)k0 * HID;
        v2f vbl; vbl.x = bl[0]; vbl.y = bl[HID];
        v2f vbr; vbr.x = br[0]; vbr.y = br[HID];
        c = wmma_k4(am, vbl, c);
        c = wmma_k4(ah, vbr, c);
    }

    const float bn = bias[nb + l15];
    float* orow = hout + (size_t)(blockIdx.x * 16 + 8 * half) * HID + nb + l15;
#pragma unroll
    for (int v = 0; v < 8; ++v) {
        float val = c[v] + bn;
        orow[(size_t)v * HID] = fmaxf(val, 0.0f);
    }
}

// out[16x64 tile] += h_l(16x128) @ fc_w_l(128x64)   (JK-cat fused as accumulation)
// block = 128 (4 waves); wave w owns out col tile w.
__global__ void __launch_bounds__(128)
k_out(const float* __restrict__ h, const float* __restrict__ fcw,
      float* __restrict__ out) {
    const int w    = threadIdx.x >> 5;   // 0..3
    const int lane = threadIdx.x & 31;
    const int half = lane >> 4;
    const int l15  = lane & 15;
    const int nb   = w * 16;

    const float* hrow = h + (size_t)(blockIdx.x * 16 + l15) * HID + 2 * half;
    const float* bp   = fcw + (size_t)(2 * half) * OUTD + nb + l15;
    float* obase      = out + (size_t)(blockIdx.x * 16 + 8 * half) * OUTD + nb + l15;

    v8f c;
#pragma unroll
    for (int v = 0; v < 8; ++v) c[v] = obase[(size_t)v * OUTD];

    for (int k0 = 0; k0 < HID; k0 += 4) {
        float2 t = *reinterpret_cast<const float2*>(hrow + k0);
        v2f a; a.x = t.x; a.y = t.y;
        const float* p = bp + (size_t)k0 * OUTD;
        v2f b; b.x = p[0]; b.y = p[OUTD];
        c = wmma_k4(a, b, c);
    }
#pragma unroll
    for (int v = 0; v < 8; ++v) obase[(size_t)v * OUTD] = c[v];
}

// ---------------- host launcher ----------------

extern "C" void kernel_launch(void* const* d_in, const int* in_sizes, int n_in,
                              void* d_out, int out_size, void* d_ws, size_t ws_size,
                              hipStream_t stream) {
    const float* x   = (const float*)d_in[0];
    const int*   ei  = (const int*)d_in[1];
    const float* Wl  = (const float*)d_in[2];
    const float* Wr  = (const float*)d_in[3];
    const float* b   = (const float*)d_in[4];
    const float* fcw = (const float*)d_in[5];
    const float* fcb = (const float*)d_in[6];
    float* out = (float*)d_out;

    const int N = in_sizes[0] / HID;   // 100000 (divisible by 16)
    const int E = in_sizes[1] / 2;     // 1600000
    const int* src = ei;
    const int* dst = ei + E;

    char* ws = (char*)d_ws;
    const size_t hbytes = (size_t)N * HID * sizeof(float);
    float* agg = (float*)(ws);                 // 51.2 MB, L2-resident during scatter
    float* hA  = (float*)(ws + hbytes);
    float* hB  = (float*)(ws + 2 * hbytes);
    float* deg = (float*)(ws + 3 * hbytes);    // 400 KB

    // init: deg = 0, out = fc_b
    k_zero<<<(N + 255) / 256, 256, 0, stream>>>(deg, N);
    {
        int total = N * OUTD;
        k_init_out<<<(total + 255) / 256, 256, 0, stream>>>(out, fcb, total);
    }
    k_degree<<<(E + 255) / 256, 256, 0, stream>>>(dst, deg, E);

    const int mtiles = N / 16;
    const float* hin = x;
    float* houts[NLAYERS] = { hA, hB, hA };

    for (int l = 0; l < NLAYERS; ++l) {
        const int z4 = N * HID / 4;
        k_zero4<<<(z4 + 255) / 256, 256, 0, stream>>>((float4*)agg, z4);

        const long long sthreads = (long long)E * 32;
        k_scatter<<<(int)((sthreads + 255) / 256), 256, 0, stream>>>(hin, src, dst, agg, E);

        k_sage<<<mtiles, 256, 0, stream>>>(hin, agg, deg,
                                           Wl + (size_t)l * HID * HID,
                                           Wr + (size_t)l * HID * HID,
                                           b + (size_t)l * HID,
                                           houts[l]);

        k_out<<<mtiles, 128, 0, stream>>>(houts[l], fcw + (size_t)l * HID * OUTD, out);

        hin = houts[l];
    }
}